// GraphSAGE_36026185678961
// MI455X (gfx1250) — compile-verified
//
#include <hip/hip_runtime.h>
#include <hip/hip_bf16.h>

typedef __attribute__((ext_vector_type(2))) float v2f;
typedef __attribute__((ext_vector_type(8))) float v8f;

#define LN_EPS 1e-5f

// ---------------------------------------------------------------- utilities
__global__ void fill_f32(float* __restrict__ p, float v, int n) {
    int i = blockIdx.x * blockDim.x + threadIdx.x;
    if (i < n) p[i] = v;
}

__global__ void count_edges(const int* __restrict__ dst, float* __restrict__ cnt, int E) {
    int e = blockIdx.x * blockDim.x + threadIdx.x;
    if (e < E) atomicAdd(&cnt[dst[e]], 1.0f);
}

__global__ void finalize_invdeg(float* __restrict__ deg, int N) {
    int i = blockIdx.x * blockDim.x + threadIdx.x;
    if (i < N) deg[i] = 1.0f / fmaxf(deg[i], 1.0f);
}

// ------------------------------------------------------- WMMA dual GEMM
// t = xin @ Wl^T          (scattered later, then scaled by invdeg)
// u = xin @ Wr^T + bl     (the "self" path + bias)
// One wave computes one 16x16 output tile using V_WMMA_F32_16X16X4_F32,
// sharing the A fragment between the Wl and Wr accumulators.
__global__ void __launch_bounds__(256)
sage_gemm(const float* __restrict__ xin,
          const float* __restrict__ Wl, const float* __restrict__ Wr,
          const float* __restrict__ bl,
          float* __restrict__ t, float* __restrict__ u,
          int N, int in_dim, int out_dim)
{
    extern __shared__ float sx[];
    const int ncol  = out_dim >> 4;     // # of 16-wide col tiles (4 or 8)
    const int rows  = (8 / ncol) << 4;  // rows per block (16 or 32)
    const int ldx   = in_dim + 4;       // pad: stride 132 -> conflict-free banks
    const int rowBase = blockIdx.x * rows;

    // cooperative stage of the A tile (rows x in_dim) into LDS
    for (int idx = threadIdx.x; idx < rows * in_dim; idx += 256) {
        int r = idx / in_dim, k = idx - r * in_dim;
        int gr = rowBase + r;
        sx[r * ldx + k] = (gr < N) ? xin[(size_t)gr * in_dim + k] : 0.0f;
    }
    __syncthreads();

    const int wave = threadIdx.x >> 5;
    const int ln   = threadIdx.x & 31;
    const int half = ln >> 4;           // K-pair select (ISA A/B layout)
    const int m    = ln & 15;           // M for A, N for B/C/D
    const int rt   = wave / ncol;
    const int ct   = wave - rt * ncol;
    const int col  = ct * 16 + m;

    const float* xrow = &sx[(rt * 16 + m) * ldx + 2 * half];
    const float* wlp  = Wl + (size_t)col * in_dim + 2 * half;
    const float* wrp  = Wr + (size_t)col * in_dim + 2 * half;

    v8f accL = {};
    v8f accR = {};
    for (int k = 0; k < in_dim; k += 4) {
        v2f a;                              // A[m][k+2h], A[m][k+2h+1]
        a.x = xrow[k];
        a.y = xrow[k + 1];
        float2 wl2 = *(const float2*)(wlp + k);
        float2 wr2 = *(const float2*)(wrp + k);
        v2f bL; bL.x = wl2.x; bL.y = wl2.y; // B[k+2h][col], B[k+2h+1][col]
        v2f bR; bR.x = wr2.x; bR.y = wr2.y;
        accL = __builtin_amdgcn_wmma_f32_16x16x4_f32(
            false, a, false, bL, (short)0, accL, false, false);
        accR = __builtin_amdgcn_wmma_f32_16x16x4_f32(
            false, a, false, bR, (short)0, accR, false, false);
    }

    const float bias = bl[col];
    const int r0 = rowBase + rt * 16 + half * 8;   // D layout: VGPR j -> M=j / M=8+j
    for (int j = 0; j < 8; ++j) {
        int r = r0 + j;
        if (r < N) {
            t[(size_t)r * out_dim + col] = accL[j];
            u[(size_t)r * out_dim + col] = accR[j] + bias;
        }
    }
}

// ------------------------------------------------------- edge scatter (sum)
__global__ void scatter_add(const float* __restrict__ t,
                            const int* __restrict__ src, const int* __restrict__ dst,
                            float* __restrict__ agg, int E, int out_dim)
{
    const int q4 = out_dim >> 2;
    long long idx = (long long)blockIdx.x * blockDim.x + threadIdx.x;
    if (idx >= (long long)E * q4) return;
    int e = (int)(idx / q4);
    int q = (int)(idx - (long long)e * q4);
    int s = src[e], d = dst[e];
    const float4 v = *(const float4*)(t + (size_t)s * out_dim + (q << 2));
    float* o = agg + (size_t)d * out_dim + (q << 2);
    atomicAdd(o + 0, v.x);
    atomicAdd(o + 1, v.y);
    atomicAdd(o + 2, v.z);
    atomicAdd(o + 3, v.w);
}

// --------------------------------------- fused mean-scale + LayerNorm + ReLU
__global__ void __launch_bounds__(256)
combine_ln(const float* __restrict__ agg, const float* __restrict__ u,
           const float* __restrict__ invdeg,
           const float* __restrict__ gamma, const float* __restrict__ beta,
           float* __restrict__ xout, int N, int out_dim)
{
    const int wave = threadIdx.x >> 5;
    const int ln   = threadIdx.x & 31;
    const int r    = blockIdx.x * 8 + wave;
    if (r >= N) return;

    const float inv = invdeg[r];
    const int ne = out_dim >> 5;    // elems per lane: 4 (128) or 2 (64)
    float vals[4];
    float s = 0.f;
    for (int i = 0; i < ne; ++i) {
        int o = ln + (i << 5);
        float v = agg[(size_t)r * out_dim + o] * inv + u[(size_t)r * out_dim + o];
        vals[i] = v;
        s += v;
    }
    for (int off = 16; off; off >>= 1) s += __shfl_xor(s, off, 32);
    const float mu = s / (float)out_dim;

    float ss = 0.f;
    for (int i = 0; i < ne; ++i) { float d = vals[i] - mu; ss += d * d; }
    for (int off = 16; off; off >>= 1) ss += __shfl_xor(ss, off, 32);
    const float rstd = rsqrtf(ss / (float)out_dim + LN_EPS);

    for (int i = 0; i < ne; ++i) {
        int o = ln + (i << 5);
        float y = (vals[i] - mu) * rstd * gamma[o] + beta[o];
        xout[(size_t)r * out_dim + o] = fmaxf(y, 0.f);
    }
}

// ------------------------------------------------------------- graph pooling
// pool[g][0:OUT] = sum (later /= count); pool[g][OUT:2*OUT] = max.
// Post-ReLU values are >= 0, so uint-bitwise atomicMax == float max.
__global__ void pool_scatter(const float* __restrict__ h, const int* __restrict__ batch,
                             float* __restrict__ pool, float* __restrict__ gcnt,
                             int N, int OUT)
{
    long long idx = (long long)blockIdx.x * blockDim.x + threadIdx.x;
    if (idx >= (long long)N * OUT) return;
    int n = (int)(idx / OUT), f = (int)(idx - (long long)n * OUT);
    int b = batch[n];
    float v = h[(size_t)n * OUT + f];
    atomicAdd(&pool[(size_t)b * 2 * OUT + f], v);
    atomicMax((unsigned int*)&pool[(size_t)b * 2 * OUT + OUT + f], __float_as_uint(v));
    if (f == 0) atomicAdd(&gcnt[b], 1.0f);
}

__global__ void pool_finalize(float* __restrict__ pool, const float* __restrict__ gcnt,
                              int G, int OUT)
{
    int idx = blockIdx.x * blockDim.x + threadIdx.x;
    if (idx >= G * OUT) return;
    int g = idx / OUT, f = idx - g * OUT;
    pool[(size_t)g * 2 * OUT + f] /= fmaxf(gcnt[g], 1.0f);
}

// ------------------------------------------------------------- tiny MLP head
__global__ void __launch_bounds__(256)
classifier(const float* __restrict__ pool, const float* __restrict__ cW1,
           const float* __restrict__ cb1, const float* __restrict__ cW2,
           const float* __restrict__ cb2, float* __restrict__ out,
           int G, int OUT, int H, int C)
{
    __shared__ float sz[64 * 128];
    const int IN2 = 2 * OUT;
    for (int idx = threadIdx.x; idx < G * H; idx += 256) {
        int g = idx / H, o = idx - g * H;
        float dot = cb1[o];
        const float* pr = pool + (size_t)g * IN2;
        const float* wr = cW1 + (size_t)o * IN2;
        for (int k = 0; k < IN2; ++k) dot += pr[k] * wr[k];
        sz[idx] = fmaxf(dot, 0.f);
    }
    __syncthreads();
    for (int idx = threadIdx.x; idx < G * C; idx += 256) {
        int g = idx / C, c = idx - g * C;
        float dot = cb2[c];
        for (int k = 0; k < H; ++k) dot += sz[g * H + k] * cW2[(size_t)c * H + k];
        out[idx] = dot;
    }
}

// ---------------------------------------------------------------- launcher
extern "C" void kernel_launch(void* const* d_in, const int* in_sizes, int n_in,
                              void* d_out, int out_size, void* d_ws, size_t ws_size,
                              hipStream_t stream)
{
    const float* x   = (const float*)d_in[0];
    const int*   ei  = (const int*)d_in[1];
    const int*   bat = (const int*)d_in[2];
    const float* Wl0 = (const float*)d_in[3];
    const float* bl0 = (const float*)d_in[4];
    const float* Wr0 = (const float*)d_in[5];
    const float* g0  = (const float*)d_in[6];
    const float* be0 = (const float*)d_in[7];
    const float* Wl1 = (const float*)d_in[8];
    const float* bl1 = (const float*)d_in[9];
    const float* Wr1 = (const float*)d_in[10];
    const float* g1  = (const float*)d_in[11];
    const float* be1 = (const float*)d_in[12];
    const float* Wl2 = (const float*)d_in[13];
    const float* bl2 = (const float*)d_in[14];
    const float* Wr2 = (const float*)d_in[15];
    const float* g2  = (const float*)d_in[16];
    const float* be2 = (const float*)d_in[17];
    const float* cW1 = (const float*)d_in[18];
    const float* cb1 = (const float*)d_in[19];
    const float* cW2 = (const float*)d_in[20];
    const float* cb2 = (const float*)d_in[21];

    const int N   = in_sizes[2];
    const int E   = in_sizes[1] / 2;
    const int IN  = in_sizes[0] / N;
    const int H   = in_sizes[4];
    const int OUT = in_sizes[14];
    const int C   = in_sizes[21];
    const int G   = out_size / C;

    float* ws = (float*)d_ws;
    const size_t nf = (size_t)N * H;
    float* T0   = ws;
    float* T1   = T0 + nf;
    float* U    = T1 + nf;
    float* AG   = U + nf;
    float* DEG  = AG + nf;
    float* POOL = DEG + N;
    float* GCNT = POOL + (size_t)G * 2 * OUT;
    const int* srcI = ei;
    const int* dstI = ei + E;

    auto cdiv = [](long long a, long long b) { return (int)((a + b - 1) / b); };

    // per-dst inverse degree (graph is constant across layers -> compute once)
    fill_f32<<<cdiv(N, 256), 256, 0, stream>>>(DEG, 0.f, N);
    count_edges<<<cdiv(E, 256), 256, 0, stream>>>(dstI, DEG, E);
    finalize_invdeg<<<cdiv(N, 256), 256, 0, stream>>>(DEG, N);

    struct Layer {
        const float* xin; const float* Wl; const float* bl; const float* Wr;
        const float* g; const float* be; float* t; float* xout; int in; int out;
    };
    Layer L[3] = {
        { x,  Wl0, bl0, Wr0, g0, be0, T0, T0, IN, H   },
        { T0, Wl1, bl1, Wr1, g1, be1, T1, T1, H,  H   },
        { T1, Wl2, bl2, Wr2, g2, be2, T0, T0, H,  OUT },
    };
    for (int l = 0; l < 3; ++l) {
        const int in = L[l].in, out = L[l].out;
        const int ncol = out / 16;
        const int rows = (8 / ncol) * 16;
        const size_t lds = (size_t)rows * (in + 4) * sizeof(float);
        sage_gemm<<<cdiv(N, rows), 256, lds, stream>>>(
            L[l].xin, L[l].Wl, L[l].Wr, L[l].bl, L[l].t, U, N, in, out);
        fill_f32<<<cdiv((long long)N * out, 256), 256, 0, stream>>>(AG, 0.f, N * out);
        scatter_add<<<cdiv((long long)E * (out / 4), 256), 256, 0, stream>>>(
            L[l].t, srcI, dstI, AG, E, out);
        combine_ln<<<cdiv(N, 8), 256, 0, stream>>>(
            AG, U, DEG, L[l].g, L[l].be, L[l].xout, N, out);
    }

    fill_f32<<<cdiv(G * 2 * OUT, 256), 256, 0, stream>>>(POOL, 0.f, G * 2 * OUT);
    fill_f32<<<cdiv(G, 256), 256, 0, stream>>>(GCNT, 0.f, G);
    pool_scatter<<<cdiv((long long)N * OUT, 256), 256, 0, stream>>>(T0, bat, POOL, GCNT, N, OUT);
    pool_finalize<<<cdiv(G * OUT, 256), 256, 0, stream>>>(POOL, GCNT, G, OUT);
    classifier<<<1, 256, 0, stream>>>(POOL, cW1, cb1, cW2, cb2, (float*)d_out, G, OUT, H, C);
}